// Point_MAE_87162066305525
// MI455X (gfx1250) — compile-verified
//
#include <hip/hip_runtime.h>
#include <float.h>

#define BATCH 32
#define NPTS 8192
#define NGRP 512
#define GSZ 32

typedef __attribute__((ext_vector_type(2))) float v2f;
typedef __attribute__((ext_vector_type(8))) float v8f;

__device__ __forceinline__ void maxCombine(float& d, int& i, float od, int oi) {
  if (od > d || (od == d && oi < i)) { d = od; i = oi; }
}
__device__ __forceinline__ void minCombine(float& d, int& i, float od, int oi) {
  if (od < d || (od == d && oi < i)) { d = od; i = oi; }
}

// ---------------------------------------------------------------------------
// Kernel 1: farthest point sampling. One block per batch. 512 threads.
// Running-min distance array in LDS; xyz (3 MB total) stays hot in 192 MB L2.
// Sequential 511 dependent argmax steps; first-index tie-break = jnp.argmax.
// ---------------------------------------------------------------------------
__global__ void fps_kernel(const float* __restrict__ xyz, float* __restrict__ centers) {
  const int b   = blockIdx.x;
  const int tid = threadIdx.x;
  const int nth = blockDim.x;            // 512
  const int nwaves = nth >> 5;           // 16

  __shared__ float sd[NPTS];             // running min distance, 32 KB
  __shared__ float rD[16];
  __shared__ int   rI[16];
  __shared__ int   sLastIdx;

  const float* base = xyz + (size_t)b * NPTS * 3;

  for (int i = tid; i < NPTS; i += nth) sd[i] = 1e10f;
  if (tid == 0) {
    sLastIdx = 0;                         // first center = point 0 (reference init)
    centers[((size_t)b * NGRP + 0) * 3 + 0] = base[0];
    centers[((size_t)b * NGRP + 0) * 3 + 1] = base[1];
    centers[((size_t)b * NGRP + 0) * 3 + 2] = base[2];
  }
  __syncthreads();

  for (int s = 1; s < NGRP; ++s) {
    const int last = sLastIdx;
    const float lx = base[last * 3 + 0];  // scalar broadcast load
    const float ly = base[last * 3 + 1];
    const float lz = base[last * 3 + 2];

    float best = -1.0f; int bestI = NPTS;
    for (int i = tid; i < NPTS; i += nth) {       // ascending i -> first-index ties
      float dx = base[i * 3 + 0] - lx;
      float dy = base[i * 3 + 1] - ly;
      float dz = base[i * 3 + 2] - lz;
      float d  = dx * dx + dy * dy + dz * dz;
      float nd = fminf(sd[i], d);
      sd[i] = nd;
      if (nd > best) { best = nd; bestI = i; }
    }
    // wave32 argmax with lowest-index tie-break
    #pragma unroll
    for (int m = 16; m >= 1; m >>= 1) {
      float od = __shfl_xor(best, m, 32);
      int   oi = __shfl_xor(bestI, m, 32);
      maxCombine(best, bestI, od, oi);
    }
    if ((tid & 31) == 0) { rD[tid >> 5] = best; rI[tid >> 5] = bestI; }
    __syncthreads();
    if (tid == 0) {
      float bb = rD[0]; int bi = rI[0];
      for (int w = 1; w < nwaves; ++w) maxCombine(bb, bi, rD[w], rI[w]);
      sLastIdx = bi;
      centers[((size_t)b * NGRP + s) * 3 + 0] = base[bi * 3 + 0];
      centers[((size_t)b * NGRP + s) * 3 + 1] = base[bi * 3 + 1];
      centers[((size_t)b * NGRP + s) * 3 + 2] = base[bi * 3 + 2];
    }
    __syncthreads();
  }
}

// ---------------------------------------------------------------------------
// 16x16 squared-distance tile via V_WMMA_F32_16X16X4_F32:
//   A row m = (-2Px, -2Py, -2Pz, |P|^2), B col n = (Qx, Qy, Qz, 1),
//   C(m,n)  = |Q_n|^2   =>   D(m,n) = |P_m - Q_n|^2  (exact f32 GEMM path)
// A layout: lanes 0-15 hold K=0,1; lanes 16-31 hold K=2,3 (M = lane&15).
// B layout: same striping with N = lane&15.  C/D: lane = column, VGPR = row.
// ---------------------------------------------------------------------------
template <int R, int C>
__device__ __forceinline__ v8f dist_tile(float px, float py, float pz, float psq,
                                         float qx, float qy, float qz, float qsq,
                                         int lane) {
  const int srcA = R * 16 + (lane & 15);
  const int srcB = C * 16 + (lane & 15);
  const float apx  = __shfl(px,  srcA, 32);
  const float apy  = __shfl(py,  srcA, 32);
  const float apz  = __shfl(pz,  srcA, 32);
  const float apsq = __shfl(psq, srcA, 32);
  const float bqx  = __shfl(qx,  srcB, 32);
  const float bqy  = __shfl(qy,  srcB, 32);
  const float bqz  = __shfl(qz,  srcB, 32);
  const float bqsq = __shfl(qsq, srcB, 32);
  const bool hi = lane >= 16;
  v2f a, bvec;
  a[0]    = hi ? (-2.0f * apz) : (-2.0f * apx);
  a[1]    = hi ? apsq          : (-2.0f * apy);
  bvec[0] = hi ? bqz  : bqx;
  bvec[1] = hi ? 1.0f : bqy;
  v8f c;
  #pragma unroll
  for (int r = 0; r < 8; ++r) c[r] = bqsq;
  return __builtin_amdgcn_wmma_f32_16x16x4_f32(false, a, false, bvec,
                                               (short)0, c, false, false);
}

// ---------------------------------------------------------------------------
// Kernel 2: per (b,g): d^2 to all 8192 points -> LDS, incremental top-32
// (ascending distance, lowest-index ties = lax.top_k), then symmetric chamfer
// between pred[b,g] and center-relative neighborhood via 4 WMMA tiles.
//
// Selection is incremental: each thread caches the min over its 32 owned
// distances; after extracting a winner only the owning thread (bi % 256)
// rescans — ~30x less serial LDS traffic than a full rescan per iteration.
// 256 threads = 8 waves; chamfer computed redundantly by every wave so EXEC
// is all-ones for WMMA; only thread 0 stores.
// ---------------------------------------------------------------------------
__global__ void knn_chamfer_kernel(const float* __restrict__ xyz,
                                   const float* __restrict__ pred,
                                   const float* __restrict__ centers,
                                   float* __restrict__ gloss) {
  const int bg  = blockIdx.x;            // b*NGRP + g
  const int b   = bg / NGRP;
  const int tid = threadIdx.x;           // 256

  __shared__ float sd2[NPTS];            // 32 KB
  __shared__ int   sknn[GSZ];
  __shared__ float rD[8];
  __shared__ int   rI[8];
  __shared__ int   sWin;

  const float cx = centers[(size_t)bg * 3 + 0];
  const float cy = centers[(size_t)bg * 3 + 1];
  const float cz = centers[(size_t)bg * 3 + 2];
  const float* base = xyz + (size_t)b * NPTS * 3;

  // Phase 1: squared distances center -> all points, plus per-thread cached min
  float best = FLT_MAX; int bestI = NPTS;
  for (int i = tid; i < NPTS; i += 256) {
    float dx = base[i * 3 + 0] - cx;
    float dy = base[i * 3 + 1] - cy;
    float dz = base[i * 3 + 2] - cz;
    float d  = dx * dx + dy * dy + dz * dz;
    sd2[i] = d;
    if (d < best) { best = d; bestI = i; }   // ascending i -> lowest-index ties
  }
  __syncthreads();

  // Phase 2: 32 incremental argmin extractions
  for (int s = 0; s < GSZ; ++s) {
    float rb = best; int ri = bestI;
    #pragma unroll
    for (int m = 16; m >= 1; m >>= 1) {
      float od = __shfl_xor(rb, m, 32);
      int   oi = __shfl_xor(ri, m, 32);
      minCombine(rb, ri, od, oi);
    }
    if ((tid & 31) == 0) { rD[tid >> 5] = rb; rI[tid >> 5] = ri; }
    __syncthreads();                      // (1) rD/rI ready; prev tid0 read done
    if (tid == 0) {
      float bb = rD[0]; int bi = rI[0];
      for (int w = 1; w < 8; ++w) minCombine(bb, bi, rD[w], rI[w]);
      sknn[s] = bi;
      sd2[bi] = FLT_MAX;                  // mask extracted element
      sWin = bi;
    }
    __syncthreads();                      // (2) sWin + sd2 update visible
    if ((sWin & 255) == tid) {            // only the owner refreshes its cache
      best = FLT_MAX; bestI = NPTS;
      for (int i = tid; i < NPTS; i += 256) {
        float d = sd2[i];
        if (d < best) { best = d; bestI = i; }
      }
    }
  }

  // Phase 3: chamfer(pred, neighborhood) via WMMA. m = lane.
  const int lane = tid & 31;
  const int qi = sknn[lane];
  const float qx = base[qi * 3 + 0] - cx;
  const float qy = base[qi * 3 + 1] - cy;
  const float qz = base[qi * 3 + 2] - cz;
  const float qsq = qx * qx + qy * qy + qz * qz;
  const float* pb = pred + ((size_t)bg * GSZ + lane) * 3;
  const float px = pb[0], py = pb[1], pz = pb[2];
  const float psq = px * px + py * py + pz * pz;

  const v8f d00 = dist_tile<0, 0>(px, py, pz, psq, qx, qy, qz, qsq, lane);
  const v8f d01 = dist_tile<0, 1>(px, py, pz, psq, qx, qy, qz, qsq, lane);
  const v8f d10 = dist_tile<1, 0>(px, py, pz, psq, qx, qy, qz, qsq, lane);
  const v8f d11 = dist_tile<1, 1>(px, py, pz, psq, qx, qy, qz, qsq, lane);

  // Row mins (min over all 32 columns), then sum over 32 rows.
  float rowTotal = 0.0f;
  {
    float rs0 = 0.0f, rs1 = 0.0f;
    #pragma unroll
    for (int r = 0; r < 8; ++r) {
      float v0 = fminf(d00[r], d01[r]);   // same row, two column tiles
      float v1 = fminf(d10[r], d11[r]);
      #pragma unroll
      for (int m = 8; m >= 1; m >>= 1) {  // min within 16-lane half (same row)
        v0 = fminf(v0, __shfl_xor(v0, m, 32));
        v1 = fminf(v1, __shfl_xor(v1, m, 32));
      }
      rs0 += v0;                          // lanes<16: rows 0-7 / lanes>=16: rows 8-15
      rs1 += v1;
    }
    rowTotal = (rs0 + __shfl_xor(rs0, 16, 32)) + (rs1 + __shfl_xor(rs1, 16, 32));
  }

  // Col mins (min over all 32 rows), then sum over 32 cols.
  float colTotal = 0.0f;
  {
    float t0 = fminf(d00[0], d10[0]);     // same column, two row tiles
    float t1 = fminf(d01[0], d11[0]);
    #pragma unroll
    for (int r = 1; r < 8; ++r) {
      t0 = fminf(t0, fminf(d00[r], d10[r]));
      t1 = fminf(t1, fminf(d01[r], d11[r]));
    }
    t0 = fminf(t0, __shfl_xor(t0, 16, 32));  // merge the two row-halves
    t1 = fminf(t1, __shfl_xor(t1, 16, 32));
    #pragma unroll
    for (int m = 8; m >= 1; m >>= 1) {       // sum 16 distinct colmins per tile
      t0 += __shfl_xor(t0, m, 32);
      t1 += __shfl_xor(t1, m, 32);
    }
    colTotal = t0 + t1;
  }

  const float loss = (rowTotal + colTotal) * (1.0f / (float)GSZ);
  if (tid == 0) gloss[bg] = loss;
}

// ---------------------------------------------------------------------------
// Kernel 3: deterministic fixed-order reduction of B*G group losses -> scalar
// ---------------------------------------------------------------------------
__global__ void reduce_kernel(const float* __restrict__ gloss, float* __restrict__ out) {
  __shared__ float s[256];
  const int tid = threadIdx.x;
  float acc = 0.0f;
  for (int i = tid; i < BATCH * NGRP; i += 256) acc += gloss[i];
  s[tid] = acc;
  __syncthreads();
  for (int k = 128; k >= 1; k >>= 1) {
    if (tid < k) s[tid] += s[tid + k];
    __syncthreads();
  }
  if (tid == 0) out[0] = s[0] * (1.0f / (float)(BATCH * NGRP));
}

extern "C" void kernel_launch(void* const* d_in, const int* in_sizes, int n_in,
                              void* d_out, int out_size, void* d_ws, size_t ws_size,
                              hipStream_t stream) {
  const float* xyz  = (const float*)d_in[0];   // [32, 8192, 3] f32
  const float* pred = (const float*)d_in[1];   // [32, 512, 32, 3] f32
  float* out = (float*)d_out;                  // scalar f32

  float* centers = (float*)d_ws;               // B*G*3 floats   (192 KB)
  float* gloss   = centers + (size_t)BATCH * NGRP * 3;  // B*G floats (64 KB)

  fps_kernel<<<BATCH, 512, 0, stream>>>(xyz, centers);
  knn_chamfer_kernel<<<BATCH * NGRP, 256, 0, stream>>>(xyz, pred, centers, gloss);
  reduce_kernel<<<1, 256, 0, stream>>>(gloss, out);
}